// HConstructor9_69363721830617
// MI455X (gfx1250) — compile-verified
//
#include <hip/hip_runtime.h>
#include <hip/hip_bf16.h>
#include <math.h>

typedef float v2f __attribute__((ext_vector_type(2)));
typedef float v8f __attribute__((ext_vector_type(8)));

#define TREP 3

// ---------------------------------------------------------------------------
// Generic 64x64 GEMM via V_WMMA_F32_16X16X4_F32.
// C[m, 0..63] = (relu?)(A[m, 0..63]) @ B(64x64) * outScale + bias
// One wave -> 16 rows x 64 cols (4 accumulator tiles). Block = 8 waves = 128 rows.
// B staged in LDS (optionally transposed). Epilogue can replicate the output
// `rep` times with a row stride of `repStride` rows (for tiled all_features).
// ---------------------------------------------------------------------------
__global__ __launch_bounds__(256)
void gemm64_wmma(const float* __restrict__ A, const float* __restrict__ B,
                 const float* __restrict__ bias, float* __restrict__ C,
                 int M, int reluIn, int bTrans, float outScale,
                 int rep, long long repStride)
{
    __shared__ float Bs[64 * 64];
    const int tid = threadIdx.x;
    for (int i = tid; i < 64 * 64; i += 256) {
        const int r = i >> 6, c = i & 63;
        Bs[i] = bTrans ? B[c * 64 + r] : B[i];
    }
    __syncthreads();

    const int wave = tid >> 5;
    const int lane = tid & 31;
    const int hi   = lane >> 4;   // 0: K pair {0,1} / rows M..M+7 ; 1: K pair {2,3} / rows M+8..
    const int lm   = lane & 15;
    const long long m0 = (long long)blockIdx.x * 128 + (long long)wave * 16;
    if (m0 >= M) return;          // wave-uniform exit, EXEC stays all-ones for WMMA

    long long arow = m0 + lm;
    if (arow > (long long)M - 1) arow = (long long)M - 1;  // clamp tail reads
    const float* Arow = A + arow * 64;

    v8f acc0 = {}, acc1 = {}, acc2 = {}, acc3 = {};
#pragma unroll
    for (int kk = 0; kk < 16; ++kk) {
        const int k2 = kk * 4 + hi * 2;
        float a0 = Arow[k2];
        float a1 = Arow[k2 + 1];
        if (reluIn) { a0 = fmaxf(a0, 0.0f); a1 = fmaxf(a1, 0.0f); }
        v2f Af = { a0, a1 };
        const float* b0p = &Bs[k2 * 64 + lm];
        const float* b1p = &Bs[(k2 + 1) * 64 + lm];
        v2f Bf;
        Bf.x = b0p[0];  Bf.y = b1p[0];
        acc0 = __builtin_amdgcn_wmma_f32_16x16x4_f32(false, Af, false, Bf, (short)0, acc0, false, false);
        Bf.x = b0p[16]; Bf.y = b1p[16];
        acc1 = __builtin_amdgcn_wmma_f32_16x16x4_f32(false, Af, false, Bf, (short)0, acc1, false, false);
        Bf.x = b0p[32]; Bf.y = b1p[32];
        acc2 = __builtin_amdgcn_wmma_f32_16x16x4_f32(false, Af, false, Bf, (short)0, acc2, false, false);
        Bf.x = b0p[48]; Bf.y = b1p[48];
        acc3 = __builtin_amdgcn_wmma_f32_16x16x4_f32(false, Af, false, Bf, (short)0, acc3, false, false);
    }

#pragma unroll
    for (int r = 0; r < 8; ++r) {
        const long long grow = m0 + r + hi * 8;      // C/D layout: vgpr r, hi half -> M = r + 8
        if (grow >= M) continue;
        float vals[4] = { acc0[r], acc1[r], acc2[r], acc3[r] };
#pragma unroll
        for (int nt = 0; nt < 4; ++nt) {
            const int col = nt * 16 + lm;
            float v = vals[nt] * outScale;
            if (bias) v += bias[col];
            const long long base = grow * 64 + col;
            for (int rp = 0; rp < rep; ++rp)
                C[base + (long long)rp * repStride * 64] = v;
        }
    }
}

// row L2 norms, eps-clamped --------------------------------------------------
__global__ void rownorm_kernel(const float* __restrict__ X, float* __restrict__ out, int rows)
{
    const int r = blockIdx.x * blockDim.x + threadIdx.x;
    if (r >= rows) return;
    const float4* p = (const float4*)(X + (long long)r * 64);
    float s = 0.0f;
#pragma unroll
    for (int i = 0; i < 16; ++i) {
        float4 v = p[i];
        s += v.x * v.x + v.y * v.y + v.z * v.z + v.w * v.w;
    }
    out[r] = fmaxf(sqrtf(s), 1e-8f);
}

// cosine-similarity edge mask; one wave per edge -----------------------------
__global__ __launch_bounds__(256)
void mask_kernel(const int* __restrict__ src, const int* __restrict__ dst,
                 const float* __restrict__ feat, const float* __restrict__ trans,
                 const float* __restrict__ nf, const float* __restrict__ ntr,
                 float* __restrict__ maskb, int E, int N)
{
    const long long e = (long long)blockIdx.x * 8 + (threadIdx.x >> 5);
    const int lane = threadIdx.x & 31;
    if (e >= E) return;
    const int s = src[e], d = dst[e];
    const float* fd = feat + (long long)d * 64;
    const float* fs = feat + (long long)s * 64;
    const float fd0 = fd[2 * lane], fd1 = fd[2 * lane + 1];
    float d0 = fs[2 * lane] * fd0 + fs[2 * lane + 1] * fd1;
    float t0, t1, t2;
    {
        const float* p = trans + ((long long)0 * N + s) * 64;
        t0 = p[2 * lane] * fd0 + p[2 * lane + 1] * fd1;
        p = trans + ((long long)1 * N + s) * 64;
        t1 = p[2 * lane] * fd0 + p[2 * lane + 1] * fd1;
        p = trans + ((long long)2 * N + s) * 64;
        t2 = p[2 * lane] * fd0 + p[2 * lane + 1] * fd1;
    }
#pragma unroll
    for (int off = 16; off >= 1; off >>= 1) {
        d0 += __shfl_xor(d0, off, 32);
        t0 += __shfl_xor(t0, off, 32);
        t1 += __shfl_xor(t1, off, 32);
        t2 += __shfl_xor(t2, off, 32);
    }
    if (lane == 0) {
        const float nd = nf[d];
        const float so = d0 / (nf[s] * nd);
        maskb[(long long)0 * E + e] = (t0 / (ntr[(long long)0 * N + s] * nd) > so) ? 1.0f : 0.0f;
        maskb[(long long)1 * E + e] = (t1 / (ntr[(long long)1 * N + s] * nd) > so) ? 1.0f : 0.0f;
        maskb[(long long)2 * E + e] = (t2 / (ntr[(long long)2 * N + s] * nd) > so) ? 1.0f : 0.0f;
    }
}

// degree init: self loop (+1) everywhere; identity edge (+1) for copies 1..3 -
__global__ void deg_init_kernel(float* __restrict__ deg, int N, int ntot)
{
    const int v = blockIdx.x * blockDim.x + threadIdx.x;
    if (v < ntot) deg[v] = (v < N) ? 1.0f : 2.0f;
}

__global__ void deg_edges_kernel(const int* __restrict__ dst, const float* __restrict__ maskb,
                                 float* __restrict__ deg, int E, int N)
{
    const long long id = (long long)blockIdx.x * blockDim.x + threadIdx.x;
    const long long tot = (long long)(TREP + 1) * E;
    if (id >= tot) return;
    if (id < E) {
        atomicAdd(&deg[dst[id]], 1.0f);
    } else {
        const long long k = id - E;
        const int t = (int)(k / E);
        const long long e = k - (long long)t * E;
        const float w = maskb[k];
        if (w != 0.0f) atomicAdd(&deg[dst[e] + (long long)(t + 1) * N], w);
    }
}

__global__ void dinv_kernel(float* __restrict__ deg, int ntot)
{
    const int v = blockIdx.x * blockDim.x + threadIdx.x;
    if (v < ntot) deg[v] = 1.0f / sqrtf(deg[v]);
}

// out[v][c] = b[c] + xw[v mod xwRows][c] * dinv[v]^2 ------------------------
__global__ void gcn_self_init_kernel(const float* __restrict__ xw, const float* __restrict__ bias,
                                     const float* __restrict__ dinv, float* __restrict__ out,
                                     int ntot, int xwRows)
{
    const long long idx = (long long)blockIdx.x * blockDim.x + threadIdx.x;
    if (idx >= (long long)ntot * 64) return;
    const int v = (int)(idx >> 6);
    const int c = (int)(idx & 63);
    const int xr = v % xwRows;
    const float di = dinv[v];
    out[idx] = bias[c] + xw[(long long)xr * 64 + c] * di * di;
}

// edge aggregation with float atomics; one wave per flat edge id ------------
__global__ __launch_bounds__(256)
void gcn_agg_kernel(const int* __restrict__ src, const int* __restrict__ dst,
                    const float* __restrict__ maskb, const float* __restrict__ dinv,
                    const float* __restrict__ xw, float* __restrict__ out, int E, int N)
{
    const long long wid = (long long)blockIdx.x * 8 + (threadIdx.x >> 5);
    const int lane = threadIdx.x & 31;
    const long long totE = (long long)(TREP + 1) * E + (long long)TREP * N;
    if (wid >= totE) return;
    int s, d; float w;
    if (wid < E) {
        s = src[wid]; d = dst[wid]; w = 1.0f;
    } else if (wid < (long long)(TREP + 1) * E) {
        const long long k = wid - E;
        const int t = (int)(k / E);
        const long long e = k - (long long)t * E;
        w = maskb[k];
        s = src[e]; d = dst[e] + (t + 1) * N;
    } else {
        const long long k = wid - (long long)(TREP + 1) * E;
        const int t = (int)(k / N);
        const int v = (int)(k - (long long)t * N);
        s = v; d = v + (t + 1) * N; w = 1.0f;
    }
    if (w == 0.0f) return;
    const float nrm = dinv[s] * w * dinv[d];
    const float* xr = xw + (long long)s * 64;
    float* op = out + (long long)d * 64 + 2 * lane;
    atomicAdd(op,     xr[2 * lane]     * nrm);
    atomicAdd(op + 1, xr[2 * lane + 1] * nrm);
}

// per-row argmax over 64 logits (lowest index wins ties); one wave per row --
__global__ __launch_bounds__(256)
void argmax_kernel(const float* __restrict__ logits, int* __restrict__ cls, long long rows)
{
    const long long r = (long long)blockIdx.x * 8 + (threadIdx.x >> 5);
    const int lane = threadIdx.x & 31;
    if (r >= rows) return;
    const float v0 = logits[r * 64 + 2 * lane];
    const float v1 = logits[r * 64 + 2 * lane + 1];
    float bv = v0; int bi = 2 * lane;
    if (v1 > bv) { bv = v1; bi = 2 * lane + 1; }
#pragma unroll
    for (int off = 16; off >= 1; off >>= 1) {
        const float ov = __shfl_xor(bv, off, 32);
        const int   oi = __shfl_xor(bi, off, 32);
        if (ov > bv || (ov == bv && oi < bi)) { bv = ov; bi = oi; }
    }
    if (lane == 0) cls[r] = bi;
}

// H counts: one thread per (copy, node) -------------------------------------
__global__ void hcount_kernel(const int* __restrict__ cls, float* __restrict__ H, int N)
{
    const long long idx = (long long)blockIdx.x * blockDim.x + threadIdx.x;
    if (idx >= (long long)(TREP + 1) * N) return;
    const long long n = idx % N;
    atomicAdd(&H[n * 64 + cls[idx]], 1.0f);
}

// hyper[j] += x2[n] for each distinct class j among node n's 4 copies -------
__global__ __launch_bounds__(256)
void hyper_kernel(const int* __restrict__ cls, const float* __restrict__ x2,
                  float* __restrict__ hyper, int N)
{
    const long long n = (long long)blockIdx.x * 8 + (threadIdx.x >> 5);
    const int lane = threadIdx.x & 31;
    if (n >= N) return;
    int cs[4];
#pragma unroll
    for (int c = 0; c < 4; ++c) cs[c] = cls[(long long)c * N + n];
    const float v0 = x2[n * 64 + 2 * lane];
    const float v1 = x2[n * 64 + 2 * lane + 1];
#pragma unroll
    for (int i = 0; i < 4; ++i) {
        bool dup = false;
        for (int j = 0; j < i; ++j) dup = dup || (cs[j] == cs[i]);
        if (!dup) {
            atomicAdd(&hyper[(long long)cs[i] * 64 + 2 * lane],     v0);
            atomicAdd(&hyper[(long long)cs[i] * 64 + 2 * lane + 1], v1);
        }
    }
}

// column sums of exp(H - 4) via LDS bins ------------------------------------
__global__ __launch_bounds__(256)
void colsum_kernel(const float* __restrict__ H, float* __restrict__ colsum, long long total)
{
    __shared__ float sb[64];
    if (threadIdx.x < 64) sb[threadIdx.x] = 0.0f;
    __syncthreads();
    for (long long idx = (long long)blockIdx.x * blockDim.x + threadIdx.x;
         idx < total; idx += (long long)gridDim.x * blockDim.x) {
        atomicAdd(&sb[(int)(idx & 63)], expf(H[idx] - 4.0f));
    }
    __syncthreads();
    if (threadIdx.x < 64) atomicAdd(&colsum[threadIdx.x], sb[threadIdx.x]);
}

__global__ void hsoft_kernel(float* __restrict__ H, const float* __restrict__ colsum, long long total)
{
    const long long idx = (long long)blockIdx.x * blockDim.x + threadIdx.x;
    if (idx >= total) return;
    H[idx] = expf(H[idx] - 4.0f) / colsum[(int)(idx & 63)];
}

// ---------------------------------------------------------------------------
extern "C" void kernel_launch(void* const* d_in, const int* in_sizes, int n_in,
                              void* d_out, int out_size, void* d_ws, size_t ws_size,
                              hipStream_t stream)
{
    (void)n_in; (void)out_size; (void)ws_size;
    const int*   edge_index = (const int*)d_in[0];
    const float* features   = (const float*)d_in[1];
    const float* W_lin      = (const float*)d_in[2];
    const float* b_lin      = (const float*)d_in[3];
    const float* gcn0_W     = (const float*)d_in[4];
    const float* gcn0_b     = (const float*)d_in[5];
    const float* gcn1_W     = (const float*)d_in[6];
    const float* gcn1_b     = (const float*)d_in[7];
    const float* lin1_W     = (const float*)d_in[8];
    const float* lin1_b     = (const float*)d_in[9];

    const int E    = in_sizes[0] / 2;
    const int N    = in_sizes[1] / 64;
    const int T    = TREP;
    const int ntot = (T + 1) * N;
    const int* src = edge_index;
    const int* dst = edge_index + E;
    const float SCALE = 0.125f;   // 64^-0.5

    // workspace carve (256B aligned)
    char* wsp = (char*)d_ws;
    auto wsalloc = [&](size_t nbytes) -> char* {
        char* p = wsp;
        wsp += (nbytes + 255) & ~(size_t)255;
        return p;
    };
    float* trans  = (float*)wsalloc((size_t)T * N * 64 * 4);
    float* nf     = (float*)wsalloc((size_t)N * 4);
    float* ntr    = (float*)wsalloc((size_t)T * N * 4);
    float* maskb  = (float*)wsalloc((size_t)T * E * 4);
    float* dinv   = (float*)wsalloc((size_t)ntot * 4);
    float* xw0    = (float*)wsalloc((size_t)N * 64 * 4);
    float* bufA   = (float*)wsalloc((size_t)ntot * 64 * 4);  // x1, then x2
    float* bufB   = (float*)wsalloc((size_t)ntot * 64 * 4);  // xw1, then logits
    int*   cls    = (int*)  wsalloc((size_t)ntot * 4);
    float* colsum = (float*)wsalloc(64 * 4);

    float* Hout  = (float*)d_out;                 // (N, 64) -> H counts, then H_soft
    float* hyper = Hout + (size_t)N * 64;         // (64, 64)
    float* dots  = hyper + 64 * 64;               // (ntot, 64)

    const dim3 blk(256);
    const int gbN   = (N + 127) / 128;
    const int gbTot = (ntot + 127) / 128;

    // 1) per-replica linear transforms: trans[t] = features @ W_lin[t] + b_lin[t]
    for (int t = 0; t < T; ++t)
        gemm64_wmma<<<gbN, blk, 0, stream>>>(features, W_lin + (size_t)t * 4096,
                                             b_lin + (size_t)t * 64,
                                             trans + (size_t)t * N * 64,
                                             N, 0, 0, 1.0f, 1, 0);

    // 2) eps-clamped row norms
    rownorm_kernel<<<(N + 255) / 256, blk, 0, stream>>>(features, nf, N);
    rownorm_kernel<<<(T * N + 255) / 256, blk, 0, stream>>>(trans, ntr, T * N);

    // 3) cosine-similarity edge masks
    mask_kernel<<<(E + 7) / 8, blk, 0, stream>>>(src, dst, features, trans, nf, ntr, maskb, E, N);

    // 4) degrees -> dinv (shared by both GCN layers)
    deg_init_kernel<<<(ntot + 255) / 256, blk, 0, stream>>>(dinv, N, ntot);
    {
        const long long tot = (long long)(T + 1) * E;
        deg_edges_kernel<<<(int)((tot + 255) / 256), blk, 0, stream>>>(dst, maskb, dinv, E, N);
    }
    dinv_kernel<<<(ntot + 255) / 256, blk, 0, stream>>>(dinv, ntot);

    // 5) GCN layer 0: xw0 = relu(features) @ gcn0_W (base rows only; tiles across copies)
    gemm64_wmma<<<gbN, blk, 0, stream>>>(features, gcn0_W, nullptr, xw0, N, 1, 0, 1.0f, 1, 0);
    {
        const long long tot = (long long)ntot * 64;
        gcn_self_init_kernel<<<(int)((tot + 255) / 256), blk, 0, stream>>>(xw0, gcn0_b, dinv, bufA, ntot, N);
        const long long totE = (long long)(T + 1) * E + (long long)T * N;
        gcn_agg_kernel<<<(int)((totE + 7) / 8), blk, 0, stream>>>(src, dst, maskb, dinv, xw0, bufA, E, N);
    }

    // 6) GCN layer 1: xw1 = relu(x1) @ gcn1_W (full ntot rows)
    gemm64_wmma<<<gbTot, blk, 0, stream>>>(bufA, gcn1_W, nullptr, bufB, ntot, 1, 0, 1.0f, 1, 0);
    {
        const long long tot = (long long)ntot * 64;
        gcn_self_init_kernel<<<(int)((tot + 255) / 256), blk, 0, stream>>>(bufB, gcn1_b, dinv, bufA, ntot, ntot);
        const long long totE = (long long)(T + 1) * E + (long long)T * N;
        gcn_agg_kernel<<<(int)((totE + 7) / 8), blk, 0, stream>>>(src, dst, maskb, dinv, bufB, bufA, E, N);
    }
    // bufA = x2 (all_features2)

    // 7) logits = relu(x2) @ lin1_W + lin1_b ; argmax -> classes
    gemm64_wmma<<<gbTot, blk, 0, stream>>>(bufA, lin1_W, lin1_b, bufB, ntot, 1, 0, 1.0f, 1, 0);
    argmax_kernel<<<(ntot + 7) / 8, blk, 0, stream>>>(bufB, cls, (long long)ntot);

    // 8) H counts and hyperedge features (accumulated directly in d_out)
    hipMemsetAsync(Hout, 0, (size_t)N * 64 * 4, stream);
    hipMemsetAsync(hyper, 0, (size_t)64 * 64 * 4, stream);
    hipMemsetAsync(colsum, 0, (size_t)64 * 4, stream);
    {
        const long long tot = (long long)(T + 1) * N;
        hcount_kernel<<<(int)((tot + 255) / 256), blk, 0, stream>>>(cls, Hout, N);
    }
    hyper_kernel<<<(N + 7) / 8, blk, 0, stream>>>(cls, bufA, hyper, N);

    // 9) column softmax of H (over the N axis), in place in d_out
    {
        const long long tot = (long long)N * 64;
        colsum_kernel<<<1024, blk, 0, stream>>>(Hout, colsum, tot);
        hsoft_kernel<<<(int)((tot + 255) / 256), blk, 0, stream>>>(Hout, colsum, tot);
    }

    // 10) dots = (all_features @ hyper^T) * SCALE — base N rows, replicated x4
    gemm64_wmma<<<gbN, blk, 0, stream>>>(features, hyper, nullptr, dots, N, 0, 1, SCALE, 4, (long long)N);
}